// MaxUnpool2D_58033598104126
// MI455X (gfx1250) — compile-verified
//
#include <hip/hip_runtime.h>
#include <cstdint>

// MaxUnpool2D (ENet decoder): scatter x[8,128,128,64] into out[8,256,256,64]
// at argmax indices. Reformulated as a GATHER: each output (b,ho,wo,c) has a
// unique possible source (b,ho/2,wo/2,c); emit x if mask matches, else 0.
// -> no zero-fill pass, no atomics, fully coalesced B128 NT stores.
// Roofline: 32MB x + 32MB mask reads + 128MB writes = 192MB @ 23.3TB/s ~ 8.3us.
// Input rows are staged to LDS via gfx1250 async global->LDS DMA so each
// global byte is read exactly once (4-way reuse served from LDS).

namespace {
constexpr int kB = 8, kH = 128, kW = 128, kC = 64;
constexpr int kHO = 256, kWO = 256;
constexpr int ROW  = kW * kC;    // 8192 floats per (b,h) input row
constexpr int OROW = kWO * kC;   // 16384 floats per output row
constexpr int THREADS = 1024;
}

typedef float v4f __attribute__((ext_vector_type(4)));
typedef int   v4i __attribute__((ext_vector_type(4)));

// Builtin prototype (from clang diagnostic): pointee type is v4i, with
// AS(1) source / AS(3) destination qualifiers.
typedef __attribute__((address_space(1))) v4i* g_v4i_p;
typedef __attribute__((address_space(3))) v4i* l_v4i_p;
typedef __attribute__((address_space(3))) char* l_char_p;

__device__ __forceinline__ void async_copy_b128(void* lds_ptr, const void* gptr) {
#if __has_builtin(__builtin_amdgcn_global_load_async_to_lds_b128)
  // (global src, lds dst, imm offset, imm cpol) -> global_load_async_to_lds_b128
  __builtin_amdgcn_global_load_async_to_lds_b128(
      (g_v4i_p)(uintptr_t)gptr, (l_v4i_p)lds_ptr, 0, 0);
#else
  uint32_t lds_off = (uint32_t)(uintptr_t)(l_char_p)lds_ptr;
  asm volatile("global_load_async_to_lds_b128 %0, %1, off"
               :: "v"(lds_off), "v"(gptr) : "memory");
#endif
}

__device__ __forceinline__ void wait_asynccnt0() {
#if __has_builtin(__builtin_amdgcn_s_wait_asynccnt)
  __builtin_amdgcn_s_wait_asynccnt(0);
#else
  asm volatile("s_wait_asynccnt 0" ::: "memory");
#endif
}

__global__ __launch_bounds__(THREADS)
void maxunpool_gather_kernel(const float* __restrict__ x,
                             const int*   __restrict__ mask,
                             float*       __restrict__ out) {
  __shared__ float sx[ROW];  // 32 KB
  __shared__ int   sm[ROW];  // 32 KB  (64 KB LDS total per workgroup)

  const int bh = blockIdx.x;          // b*128 + h  (1024 workgroups)
  const int t  = (int)threadIdx.x;    // 0..1023
  const int h  = bh & (kH - 1);
  const int b  = bh >> 7;

  const float* __restrict__ xrow = x    + (size_t)bh * ROW;
  const int*   __restrict__ mrow = mask + (size_t)bh * ROW;

  // Stage one input row (+mask row) into LDS: 8192 floats each = 2048 B128
  // transfers per array; 1024 lanes x 2 iterations. ASYNCcnt-tracked DMA.
#pragma unroll
  for (int i = 0; i < 2; ++i) {
    const int j = (t + i * THREADS) * 4;   // float index, 16B granule
    async_copy_b128(&sx[j], &xrow[j]);
    async_copy_b128(&sm[j], &mrow[j]);
  }
  wait_asynccnt0();   // our async writes to LDS landed
  __syncthreads();    // cross-wave visibility within the workgroup

  // Produce output rows ho = 2h and 2h+1: 2*256*64 = 32768 floats
  // = 8192 float4 results; 1024 threads -> 8 iterations.
  float* __restrict__ obase =
      out + (size_t)b * ((size_t)kHO * OROW) + (size_t)(2 * h) * OROW;

#pragma unroll
  for (int i = 0; i < 8; ++i) {
    const int p  = t + i * THREADS;        // 0..8191
    const int c  = (p & 15) << 2;          // channel (multiple of 4)
    const int wo = (p >> 4) & (kWO - 1);   // output column
    const int r  = p >> 12;                // 0 or 1 -> ho = 2h + r
    const int w  = wo >> 1;                // source column
    const int li = (w << 6) + c;           // LDS float index (16B aligned)

    const v4f v = *reinterpret_cast<const v4f*>(&sx[li]);
    const v4i q = *reinterpret_cast<const v4i*>(&sm[li]);

    // TF argmax flat layout: (ho*WOUT + wo)*C + c
    const int o = ((((2 * h + r) << 8) + wo) << 6) + c;

    v4f res;
    res.x = (q.x == o    ) ? v.x : 0.0f;
    res.y = (q.y == o + 1) ? v.y : 0.0f;
    res.z = (q.z == o + 2) ? v.z : 0.0f;
    res.w = (q.w == o + 3) ? v.w : 0.0f;

    float* dst = obase + (size_t)r * OROW + ((size_t)wo << 6) + c;
    // Write-once output: non-temporal hint keeps x/mask streams resident in L2.
    __builtin_nontemporal_store(res, reinterpret_cast<v4f*>(dst));
  }
}

extern "C" void kernel_launch(void* const* d_in, const int* in_sizes, int n_in,
                              void* d_out, int out_size, void* d_ws, size_t ws_size,
                              hipStream_t stream) {
  (void)in_sizes; (void)n_in; (void)out_size; (void)d_ws; (void)ws_size;
  const float* x    = (const float*)d_in[0];
  const int*   mask = (const int*)d_in[1];
  float*       out  = (float*)d_out;

  dim3 grid(kB * kH);      // 1024 workgroups, one per (b, h) input row
  dim3 block(THREADS);     // 32 wave32s
  hipLaunchKernelGGL(maxunpool_gather_kernel, grid, block, 0, stream,
                     x, mask, out);
}